// SimpleStateSpaceModel_66142496358539
// MI455X (gfx1250) — compile-verified
//
#include <hip/hip_runtime.h>

// ---------------------------------------------------------------------------
// Linear state-space rollout restructured as chunked scan + FP32 WMMA GEMMs.
//   state dim N=2048, obs dim M=512, T=8192 steps, chunk K=16 -> 512 chunks.
// GEMM tiles are staged into LDS with gfx1250 async global->LDS DMA
// (ASYNCcnt-tracked), double-buffered so WMMA overlaps the L2 streaming.
// ---------------------------------------------------------------------------

#define SDIM 2048
#define ODIM 512
#define TSTEPS 8192
#define KCH 16
#define NCH (TSTEPS / KCH) // 512 chunks

#define PROC_STD 0.31622776601683794f // sqrt(0.1)
#define OBS_STD 0.7071067811865476f   // sqrt(0.5)

// LDS tile strides (padded to dodge bank conflicts; multiples of 4 for b128)
#define ASTR 36              // A tile: 64 rows x 32 cols
#define BSTR 72              // B tile: 32 rows x 64 cols
#define ATILE (64 * ASTR)
#define BTILE (32 * BSTR)

typedef float v2f __attribute__((ext_vector_type(2)));
typedef float v8f __attribute__((ext_vector_type(8)));

// ---------------------------------------------------------------------------
// init: P0 = I, Wagg = 0, barrier counters = 0, scan state = initial_state,
//       chunk-start-state matrix column 0 = initial_state.
// ---------------------------------------------------------------------------
__global__ void k_init(float* __restrict__ P, float* __restrict__ Wagg,
                       int* __restrict__ cnt, float* __restrict__ svec,
                       float* __restrict__ S, const float* __restrict__ x0) {
  long idx = (long)blockIdx.x * 256 + threadIdx.x; // grid covers SDIM*SDIM
  if (idx < (long)SDIM * SDIM) {
    int r = (int)(idx / SDIM), c = (int)(idx % SDIM);
    P[idx] = (r == c) ? 1.0f : 0.0f;
  }
  if (idx < (long)SDIM * NCH) Wagg[idx] = 0.0f;
  if (idx < NCH) cnt[idx] = 0;
  if (idx < SDIM) {
    float v = x0[idx];
    svec[idx] = v;
    S[idx * NCH + 0] = v;
  }
}

// ---------------------------------------------------------------------------
// Pre-scale + re-layout process noise: Wt[i][k][c] = pn[(c*K+i)*N + k]*PROC_STD
// so every later GEMM sees a dense row-major B with contiguous chunk columns.
// ---------------------------------------------------------------------------
__global__ void k_noise_t(float* __restrict__ Wt, const float* __restrict__ pn) {
  long idx = (long)blockIdx.x * 256 + threadIdx.x; // KCH*SDIM*NCH
  int c = (int)(idx % NCH);
  long r = idx / NCH;
  int k = (int)(r % SDIM);
  int i = (int)(r / SDIM);
  Wt[idx] = pn[(long)(c * KCH + i) * SDIM + k] * PROC_STD;
}

// ---------------------------------------------------------------------------
// Issue one A(64x32) + B(32x64) tile stage as async global->LDS b128 DMA.
// 8 wave-instructions per wave (ASYNCcnt += 8). Generic LDS pointers truncate
// to the DS byte offset (aperture: addr[31:0] == LDS address).
// ---------------------------------------------------------------------------
__device__ __forceinline__ void issue_tiles(const float* __restrict__ A,
                                            int lda,
                                            const float* __restrict__ B,
                                            int ldb, int kb, int mBase,
                                            int nBase, float* As, float* Bs,
                                            int t) {
#pragma unroll
  for (int i = 0; i < 4; ++i) {
    int f = t + i * 128; // float4 slot 0..511
    // A tile: row = f/8, col = (f%8)*4
    int ra = f >> 3;
    int ca = (f & 7) * 4;
    unsigned la = (unsigned)(unsigned long long)&As[ra * ASTR + ca];
    const float* ga = &A[(long)(mBase + ra) * lda + kb + ca];
    asm volatile("global_load_async_to_lds_b128 %0, %1, off" ::"v"(la), "v"(ga)
                 : "memory");
    // B tile: row = f/16, col = (f%16)*4
    int rb = f >> 4;
    int cb = (f & 15) * 4;
    unsigned lb = (unsigned)(unsigned long long)&Bs[rb * BSTR + cb];
    const float* gb = &B[(long)(kb + rb) * ldb + nBase + cb];
    asm volatile("global_load_async_to_lds_b128 %0, %1, off" ::"v"(lb), "v"(gb)
                 : "memory");
  }
}

// ---------------------------------------------------------------------------
// 64x64 block-tile accumulation: 128 threads = 4 waves, each wave owns a
// 32x32 quadrant = 2x2 WMMA accumulators (A/B fragment reuse), K swept 32 at
// a time through double-buffered async-staged LDS tiles.
// FP32 WMMA fragment layouts per CDNA5 ISA 7.12.2:
//   A 16x4 : lane l<16 -> row m=l holds K={k0,k0+1}; l>=16 -> K={k0+2,k0+3}
//   B 4x16 : vgpr v, half h -> row K = k0+v+2h, col n = l%16
//   C 16x16: vgpr v, half h -> row M = v+8h, col n = l%16
// ---------------------------------------------------------------------------
__device__ __forceinline__ void mm_tile_db(v8f (&acc)[2][2],
                                           const float* __restrict__ A, int lda,
                                           const float* __restrict__ B, int ldb,
                                           int Kg, int mBase, int nBase,
                                           float* As, float* Bs) {
  const int t = threadIdx.x;
  const int lane = t & 31;
  const int wave = t >> 5;
  const int wm = (wave >> 1) * 32;
  const int wn = (wave & 1) * 32;
  const int lr = lane & 15;
  const int half = lane >> 4;

  issue_tiles(A, lda, B, ldb, 0, mBase, nBase, As, Bs, t);
  for (int kb = 0; kb < Kg; kb += 32) {
    const int cur = (kb >> 5) & 1;
    const float* Ac = As + cur * ATILE;
    const float* Bc = Bs + cur * BTILE;
    if (kb + 32 < Kg) {
      // Prefetch next stage into the other buffer, then drain only the
      // current stage's 8 per-wave async ops.
      issue_tiles(A, lda, B, ldb, kb + 32, mBase, nBase,
                  As + (cur ^ 1) * ATILE, Bs + (cur ^ 1) * BTILE, t);
      asm volatile("s_wait_asynccnt 0x8" ::: "memory");
    } else {
      asm volatile("s_wait_asynccnt 0x0" ::: "memory");
    }
    __syncthreads();
#pragma unroll
    for (int k0 = 0; k0 < 32; k0 += 4) {
      v2f a[2], b[2];
#pragma unroll
      for (int mi = 0; mi < 2; ++mi) {
        a[mi].x = Ac[(wm + mi * 16 + lr) * ASTR + k0 + half * 2 + 0];
        a[mi].y = Ac[(wm + mi * 16 + lr) * ASTR + k0 + half * 2 + 1];
      }
#pragma unroll
      for (int ni = 0; ni < 2; ++ni) {
        b[ni].x = Bc[(k0 + 0 + half * 2) * BSTR + wn + ni * 16 + lr];
        b[ni].y = Bc[(k0 + 1 + half * 2) * BSTR + wn + ni * 16 + lr];
      }
#pragma unroll
      for (int mi = 0; mi < 2; ++mi)
#pragma unroll
        for (int ni = 0; ni < 2; ++ni)
          acc[mi][ni] = __builtin_amdgcn_wmma_f32_16x16x4_f32(
              false, a[mi], false, b[ni], (short)0, acc[mi][ni], false, false);
    }
    __syncthreads();
  }
}

// ---------------------------------------------------------------------------
// Generic row-major FP32 WMMA GEMM: C = A@B (+C if accumulate). 64x64 tiles.
// Used for: P_{p+1} = A @ P_p ; G_p = Cobs @ P_p ; Wagg += P_p @ Wt[K-1-p].
// ---------------------------------------------------------------------------
__global__ void __launch_bounds__(128)
    k_gemm(float* __restrict__ Cc, const float* __restrict__ A,
           const float* __restrict__ B, int lda, int ldb, int ldc, int Kg,
           int accumulate) {
  __shared__ float As[2 * ATILE];
  __shared__ float Bs[2 * BTILE];
  const int mBase = blockIdx.y * 64;
  const int nBase = blockIdx.x * 64;
  v8f acc[2][2];
#pragma unroll
  for (int mi = 0; mi < 2; ++mi)
#pragma unroll
    for (int ni = 0; ni < 2; ++ni) acc[mi][ni] = (v8f){0.f, 0.f, 0.f, 0.f,
                                                       0.f, 0.f, 0.f, 0.f};
  mm_tile_db(acc, A, lda, B, ldb, Kg, mBase, nBase, As, Bs);

  const int lane = threadIdx.x & 31;
  const int wave = threadIdx.x >> 5;
  const int wm = (wave >> 1) * 32, wn = (wave & 1) * 32;
  const int lr = lane & 15, half = lane >> 4;
#pragma unroll
  for (int mi = 0; mi < 2; ++mi)
#pragma unroll
    for (int ni = 0; ni < 2; ++ni)
#pragma unroll
      for (int v = 0; v < 8; ++v) {
        long off = (long)(mBase + wm + mi * 16 + v + 8 * half) * ldc + nBase +
                   wn + ni * 16 + lr;
        float outv = acc[mi][ni][v];
        if (accumulate) outv += Cc[off];
        Cc[off] = outv;
      }
}

// ---------------------------------------------------------------------------
// Chunk-carry scan: s_{(c+1)K} = A^K s_{cK} + Wagg[:,c]. 511 sequential steps,
// persistent kernel with a per-step global arrival barrier (64 blocks).
// ---------------------------------------------------------------------------
__global__ void __launch_bounds__(256)
    k_scan(const float* __restrict__ PK, const float* __restrict__ Wagg,
           float* __restrict__ svec, float* __restrict__ S,
           int* __restrict__ cnt) {
  const int t = threadIdx.x;
  const int row = blockIdx.x * 32 + (t >> 3); // 64 blocks * 32 rows = 2048
  const int part = t & 7;
  const float* __restrict__ prow = PK + (long)row * SDIM;

  for (int c = 0; c < NCH - 1; ++c) {
    const float* sin = svec + (c & 1) * SDIM;
    float* sout = svec + ((c + 1) & 1) * SDIM;
    float sum = 0.0f;
    for (int q = 0; q < SDIM / 8; ++q) {
      int e = part + q * 8;
      sum += prow[e] * sin[e];
    }
    sum += __shfl_down(sum, 4, 8);
    sum += __shfl_down(sum, 2, 8);
    sum += __shfl_down(sum, 1, 8);
    if (part == 0) {
      float val = sum + Wagg[(long)row * NCH + c];
      sout[row] = val;
      S[(long)row * NCH + (c + 1)] = val;
    }
    // grid-wide barrier for step c
    __threadfence();
    __syncthreads();
    if (t == 0) {
      atomicAdd(&cnt[c], 1);
      while (__atomic_load_n((const int*)&cnt[c], __ATOMIC_RELAXED) <
             (int)gridDim.x)
        __builtin_amdgcn_s_sleep(1);
    }
    __syncthreads();
    __threadfence();
  }
}

// ---------------------------------------------------------------------------
// Observations (minus obs noise): for j = blockIdx.z,
//   D_j = G_j @ S  +  sum_{i<j} G_{j-1-i} @ Wt_i        (512 x 512 each)
// stored transposed D[j][c][m] via an LDS transpose so the final pass is
// fully coalesced.
// ---------------------------------------------------------------------------
__global__ void __launch_bounds__(128)
    k_obs(float* __restrict__ D, const float* __restrict__ G,
          const float* __restrict__ S, const float* __restrict__ Wt) {
  __shared__ float As[2 * ATILE];
  __shared__ float Bs[2 * BTILE];
  __shared__ float Ts[64 * 65];
  const int j = blockIdx.z;
  const int mBase = blockIdx.y * 64; // over ODIM
  const int nBase = blockIdx.x * 64; // over NCH
  v8f acc[2][2];
#pragma unroll
  for (int mi = 0; mi < 2; ++mi)
#pragma unroll
    for (int ni = 0; ni < 2; ++ni) acc[mi][ni] = (v8f){0.f, 0.f, 0.f, 0.f,
                                                       0.f, 0.f, 0.f, 0.f};

  mm_tile_db(acc, G + (long)j * ODIM * SDIM, SDIM, S, NCH, SDIM, mBase, nBase,
             As, Bs);
  for (int i = 0; i < j; ++i) {
    mm_tile_db(acc, G + (long)(j - 1 - i) * ODIM * SDIM, SDIM,
               Wt + (long)i * SDIM * NCH, NCH, SDIM, mBase, nBase, As, Bs);
  }

  const int t = threadIdx.x, lane = t & 31, wave = t >> 5;
  const int wm = (wave >> 1) * 32, wn = (wave & 1) * 32;
  const int lr = lane & 15, half = lane >> 4;
#pragma unroll
  for (int mi = 0; mi < 2; ++mi)
#pragma unroll
    for (int ni = 0; ni < 2; ++ni)
#pragma unroll
      for (int v = 0; v < 8; ++v)
        Ts[(wn + ni * 16 + lr) * 65 + wm + mi * 16 + v + 8 * half] =
            acc[mi][ni][v];
  __syncthreads();
  int cl = t >> 1;        // 0..63 local chunk col
  int m0 = (t & 1) * 32;  // 32 consecutive m per thread
  long base = ((long)j * NCH + nBase + cl) * ODIM + mBase + m0;
#pragma unroll
  for (int q = 0; q < 32; ++q) D[base + q] = Ts[cl * 65 + m0 + q];
}

// ---------------------------------------------------------------------------
// out[t, m] = D[j][c][m] + obs_noise[t, m] * OBS_STD   (t = c*K + j)
// ---------------------------------------------------------------------------
__global__ void k_finalize(float* __restrict__ out, const float* __restrict__ D,
                           const float* __restrict__ on) {
  long idx = (long)blockIdx.x * 256 + threadIdx.x; // TSTEPS*ODIM
  int m = (int)(idx % ODIM);
  long tt = idx / ODIM;
  int c = (int)(tt / KCH), j = (int)(tt % KCH);
  out[idx] = D[((long)j * NCH + c) * ODIM + m] + on[idx] * OBS_STD;
}

// ---------------------------------------------------------------------------
extern "C" void kernel_launch(void* const* d_in, const int* in_sizes, int n_in,
                              void* d_out, int out_size, void* d_ws,
                              size_t ws_size, hipStream_t stream) {
  (void)in_sizes;
  (void)n_in;
  (void)out_size;
  (void)ws_size;
  const float* x0 = (const float*)d_in[0];
  const float* A = (const float*)d_in[1];
  const float* Cm = (const float*)d_in[2];
  const float* pn = (const float*)d_in[3];
  const float* on = (const float*)d_in[4];
  float* out = (float*)d_out;

  float* ws = (float*)d_ws;
  size_t off = 0;
  float* Pping = ws + off; off += (size_t)SDIM * SDIM;           // 16 MB
  float* Ppong = ws + off; off += (size_t)SDIM * SDIM;           // 16 MB
  float* G     = ws + off; off += (size_t)KCH * ODIM * SDIM;     // 64 MB
  float* Wt    = ws + off; off += (size_t)KCH * SDIM * NCH;      // 64 MB
  float* Wagg  = ws + off; off += (size_t)SDIM * NCH;            // 4 MB
  float* S     = ws + off; off += (size_t)SDIM * NCH;            // 4 MB
  float* svec  = ws + off; off += (size_t)2 * SDIM;              // 16 KB
  float* D     = ws + off; off += (size_t)KCH * NCH * ODIM;      // 16 MB
  int* cnt = (int*)(ws + off);                                   // 2 KB

  k_init<<<(SDIM * SDIM) / 256, 256, 0, stream>>>(Pping, Wagg, cnt, svec, S,
                                                  x0);
  k_noise_t<<<(KCH * SDIM * NCH) / 256, 256, 0, stream>>>(Wt, pn);

  // Power chain: stream out G_p = C @ A^p and carry noise Wagg += A^p @ w.
  float* Pc = Pping;
  float* Pn = Ppong;
  for (int p = 0; p < KCH; ++p) {
    k_gemm<<<dim3(SDIM / 64, ODIM / 64), 128, 0, stream>>>(
        G + (size_t)p * ODIM * SDIM, Cm, Pc, SDIM, SDIM, SDIM, SDIM, 0);
    k_gemm<<<dim3(NCH / 64, SDIM / 64), 128, 0, stream>>>(
        Wagg, Pc, Wt + (size_t)(KCH - 1 - p) * SDIM * NCH, SDIM, NCH, NCH, SDIM,
        1);
    k_gemm<<<dim3(SDIM / 64, SDIM / 64), 128, 0, stream>>>(Pn, A, Pc, SDIM,
                                                           SDIM, SDIM, SDIM, 0);
    float* tmp = Pc; Pc = Pn; Pn = tmp;
  }
  // Pc now holds A^K.
  k_scan<<<64, 256, 0, stream>>>(Pc, Wagg, svec, S, cnt);
  k_obs<<<dim3(NCH / 64, ODIM / 64, KCH), 128, 0, stream>>>(D, G, S, Wt);
  k_finalize<<<(TSTEPS * ODIM) / 256, 256, 0, stream>>>(out, D, on);
}